// SimpleGCN_36747740184680
// MI455X (gfx1250) — compile-verified
//
#include <hip/hip_runtime.h>
#include <math.h>

#define N_NODES 100000
#define E_EDGES 1600000
#define DIM_IN  128
#define DIM_HID 256
#define DIM_EMB 64

typedef __attribute__((ext_vector_type(2))) float v2f;
typedef __attribute__((ext_vector_type(8))) float v8f;

// ---------------------------------------------------------------- degree prep
__global__ void k_fill1(float* __restrict__ p, int n) {
  int i = blockIdx.x * blockDim.x + threadIdx.x;
  if (i < n) p[i] = 1.0f;                    // self-loop contributes 1 to degree
}

__global__ void k_degcount(const int* __restrict__ dst, float* __restrict__ deg, int e) {
  int i = blockIdx.x * blockDim.x + threadIdx.x;
  if (i < e) unsafeAtomicAdd(&deg[dst[i]], 1.0f);
}

__global__ void k_dinv(float* __restrict__ deg, int n) {
  int i = blockIdx.x * blockDim.x + threadIdx.x;
  if (i < n) deg[i] = rsqrtf(deg[i]);        // deg >= 1 always (self loops)
}

// ---------------------------------------------------------------- fp32 WMMA GEMM
// C[M,N] = A[M,K] @ B[K,N].  One wave -> 16x64 tile via V_WMMA_F32_16X16X4_F32.
// Block = 256 threads = 8 waves -> 128 rows per block.  Wave-uniform guards only.
__global__ void k_gemm_wmma(const float* __restrict__ A, const float* __restrict__ B,
                            float* __restrict__ C, int M, int K, int N) {
  const int lane = threadIdx.x & 31;
  const int wave = threadIdx.x >> 5;
  const int row0 = (blockIdx.x * 8 + wave) * 16;
  const int col0 = blockIdx.y * 64;
  if (row0 >= M) return;                     // wave-uniform: EXEC stays all-ones

  const int hi  = lane >> 4;                 // 0: lanes 0-15, 1: lanes 16-31
  const int l16 = lane & 15;

  v8f acc0 = {}, acc1 = {}, acc2 = {}, acc3 = {};

  // A fragment: lane row = row0+l16; K components at k + 2*hi + {0,1}
  const float* Arow = A + (size_t)(row0 + l16) * K + 2 * hi;

  for (int k = 0; k < K; k += 4) {
    v2f a;
    a.x = Arow[k];
    a.y = Arow[k + 1];

    // B fragment: VGPR0 holds row k+2*hi, VGPR1 holds row k+2*hi+1, col = col0+l16
    const float* Bk0 = B + (size_t)(k + 2 * hi) * N + col0 + l16;
    const float* Bk1 = Bk0 + N;
    v2f b0, b1, b2, b3;
    b0.x = Bk0[0];  b0.y = Bk1[0];
    b1.x = Bk0[16]; b1.y = Bk1[16];
    b2.x = Bk0[32]; b2.y = Bk1[32];
    b3.x = Bk0[48]; b3.y = Bk1[48];

    acc0 = __builtin_amdgcn_wmma_f32_16x16x4_f32(false, a, false, b0, (short)0, acc0, false, false);
    acc1 = __builtin_amdgcn_wmma_f32_16x16x4_f32(false, a, false, b1, (short)0, acc1, false, false);
    acc2 = __builtin_amdgcn_wmma_f32_16x16x4_f32(false, a, false, b2, (short)0, acc2, false, false);
    acc3 = __builtin_amdgcn_wmma_f32_16x16x4_f32(false, a, false, b3, (short)0, acc3, false, false);
  }

  // C/D layout: VGPR r -> row r (lanes 0-15) / row r+8 (lanes 16-31), col = l16
  float* Cbase = C + (size_t)row0 * N + col0 + l16;
#pragma unroll
  for (int r = 0; r < 8; ++r) {
    float* p = Cbase + (size_t)(r + 8 * hi) * N;
    p[0]  = acc0[r];
    p[16] = acc1[r];
    p[32] = acc2[r];
    p[48] = acc3[r];
  }
}

// ---------------------------------------------------------------- aggregation
// agg[i,:] = T[i,:] * dinv[i]^2 + bias   (self-loop term + bias), float4 lanes
__global__ void k_agg_init(const float* __restrict__ T, const float* __restrict__ dinv,
                           const float* __restrict__ bias, float* __restrict__ agg,
                           int total4, int sh /*log2(dim/4)*/) {
  int i = blockIdx.x * blockDim.x + threadIdx.x;
  if (i >= total4) return;
  int row = i >> sh;
  int j   = (i - (row << sh)) << 2;
  float d = dinv[row];
  float s = d * d;
  const float4 t = *(const float4*)(T + ((size_t)i << 2));
  float4 r;
  r.x = t.x * s + bias[j + 0];
  r.y = t.y * s + bias[j + 1];
  r.z = t.z * s + bias[j + 2];
  r.w = t.w * s + bias[j + 3];
  *(float4*)(agg + ((size_t)i << 2)) = r;
}

// agg[dst,:] += T[src,:] * dinv[src]*dinv[dst] over all real edges.
// dim/4 consecutive lanes cover consecutive float4 chunks of one edge -> coalesced.
__global__ void k_agg_edges(const float* __restrict__ T, const float* __restrict__ dinv,
                            const int* __restrict__ src, const int* __restrict__ dst,
                            float* __restrict__ agg, int total, int sh /*log2(dim/4)*/,
                            int dim) {
  int i = blockIdx.x * blockDim.x + threadIdx.x;
  if (i >= total) return;
  int e = i >> sh;
  int j = (i & ((1 << sh) - 1)) << 2;
  int s = src[e];
  int d = dst[e];
  float nrm = dinv[s] * dinv[d];
  const float4 v = *(const float4*)(T + (size_t)s * dim + j);
  float* a = agg + (size_t)d * dim + j;
  unsafeAtomicAdd(a + 0, v.x * nrm);
  unsafeAtomicAdd(a + 1, v.y * nrm);
  unsafeAtomicAdd(a + 2, v.z * nrm);
  unsafeAtomicAdd(a + 3, v.w * nrm);
}

__global__ void k_relu(float* __restrict__ p, int n) {
  int i = blockIdx.x * blockDim.x + threadIdx.x;
  if (i < n) p[i] = fmaxf(p[i], 0.0f);
}

// ---------------------------------------------------------------- L2 row norm
// One wave per 64-wide row; wave32 shuffle reduction.
__global__ void k_l2norm(const float* __restrict__ h, float* __restrict__ out, int n) {
  int lane = threadIdx.x & 31;
  int wave = threadIdx.x >> 5;
  int row  = blockIdx.x * 8 + wave;
  if (row >= n) return;
  const float* r = h + (size_t)row * DIM_EMB;
  float a = r[lane];
  float b = r[lane + 32];
  float s = a * a + b * b;
#pragma unroll
  for (int o = 16; o > 0; o >>= 1) s += __shfl_xor(s, o, 32);
  float scale = 1.0f / fmaxf(sqrtf(s), 1e-12f);
  float* w = out + (size_t)row * DIM_EMB;
  w[lane]      = a * scale;
  w[lane + 32] = b * scale;
}

// ---------------------------------------------------------------- launcher
extern "C" void kernel_launch(void* const* d_in, const int* in_sizes, int n_in,
                              void* d_out, int out_size, void* d_ws, size_t ws_size,
                              hipStream_t stream) {
  const float* x  = (const float*)d_in[0];
  const int*   ei = (const int*)d_in[1];          // [2, E] int32
  const float* W1 = (const float*)d_in[2];
  const float* b1 = (const float*)d_in[3];
  const float* W2 = (const float*)d_in[4];
  const float* b2 = (const float*)d_in[5];
  const float* W3 = (const float*)d_in[6];
  const float* b3 = (const float*)d_in[7];
  float* out = (float*)d_out;

  float* buf0 = (float*)d_ws;                            // N x 256
  float* buf1 = buf0 + (size_t)N_NODES * DIM_HID;        // N x 256
  float* dinv = buf1 + (size_t)N_NODES * DIM_HID;        // N

  const int* src = ei;
  const int* dst = ei + E_EDGES;

  const int TB = 256;
  // ---- degrees -> dinv
  k_fill1<<<(N_NODES + TB - 1) / TB, TB, 0, stream>>>(dinv, N_NODES);
  k_degcount<<<(E_EDGES + TB - 1) / TB, TB, 0, stream>>>(dst, dinv, E_EDGES);
  k_dinv<<<(N_NODES + TB - 1) / TB, TB, 0, stream>>>(dinv, N_NODES);

  const int rowBlocks = (N_NODES + 127) / 128;

  // ---- layer 1: h1 = relu(agg(x @ W1) + b1)
  {
    dim3 g(rowBlocks, DIM_HID / 64);
    k_gemm_wmma<<<g, TB, 0, stream>>>(x, W1, buf0, N_NODES, DIM_IN, DIM_HID);
    int t4 = N_NODES * (DIM_HID / 4);
    k_agg_init<<<(t4 + TB - 1) / TB, TB, 0, stream>>>(buf0, dinv, b1, buf1, t4, 6);
    int te = E_EDGES * (DIM_HID / 4);
    k_agg_edges<<<(te + TB - 1) / TB, TB, 0, stream>>>(buf0, dinv, src, dst, buf1, te, 6, DIM_HID);
    int n = N_NODES * DIM_HID;
    k_relu<<<(n + TB - 1) / TB, TB, 0, stream>>>(buf1, n);
  }

  // ---- layer 2: h2 = relu(agg(h1 @ W2) + b2)
  {
    dim3 g(rowBlocks, DIM_HID / 64);
    k_gemm_wmma<<<g, TB, 0, stream>>>(buf1, W2, buf0, N_NODES, DIM_HID, DIM_HID);
    int t4 = N_NODES * (DIM_HID / 4);
    k_agg_init<<<(t4 + TB - 1) / TB, TB, 0, stream>>>(buf0, dinv, b2, buf1, t4, 6);
    int te = E_EDGES * (DIM_HID / 4);
    k_agg_edges<<<(te + TB - 1) / TB, TB, 0, stream>>>(buf0, dinv, src, dst, buf1, te, 6, DIM_HID);
    int n = N_NODES * DIM_HID;
    k_relu<<<(n + TB - 1) / TB, TB, 0, stream>>>(buf1, n);
  }

  // ---- layer 3: h3 = agg(h2 @ W3) + b3 ; out = l2norm(h3)
  {
    dim3 g(rowBlocks, DIM_EMB / 64);
    k_gemm_wmma<<<g, TB, 0, stream>>>(buf1, W3, buf0, N_NODES, DIM_HID, DIM_EMB);
    int t4 = N_NODES * (DIM_EMB / 4);
    k_agg_init<<<(t4 + TB - 1) / TB, TB, 0, stream>>>(buf0, dinv, b3, buf1, t4, 4);
    int te = E_EDGES * (DIM_EMB / 4);
    k_agg_edges<<<(te + TB - 1) / TB, TB, 0, stream>>>(buf0, dinv, src, dst, buf1, te, 4, DIM_EMB);
    k_l2norm<<<(N_NODES + 7) / 8, TB, 0, stream>>>(buf1, out, N_NODES);
  }
}